// TreeEncoder_59966333387027
// MI455X (gfx1250) — compile-verified
//
#include <hip/hip_runtime.h>

// TreeEncoder on MI455X (gfx1250), compile-only tuned.
//
// Key algebraic rewrite: emb[node_ids] @ W == (emb @ W)[node_ids], so the
// N=100k GEMM (13.1 GF) becomes a VOCAB=10k GEMM (1.31 GF) + row gathers.
// Everything after the GEMM is HBM-bound (~500 MB total => ~22us @ 23.3 TB/s),
// so we keep fp32 end-to-end and use the native f32 WMMA (16x16x4) for the
// matmul — exact precision at negligible cost.

typedef float v2f __attribute__((ext_vector_type(2)));
typedef float v8f __attribute__((ext_vector_type(8)));

#define EMB_DIM 256
#define ENC_DIM 256
#define ROWS_PER_BLK 512
#define MAX_GRID 256

// ---------------------------------------------------------------------------
// Kernel 1: embW[v][n] = sum_k emb[v][k] * W[k][n] + b[n]
// One wave (32 lanes) owns a 16x16 tile of embW; K advances 4 per WMMA.
// f32 WMMA operand layout (ISA 7.12.2, 32-bit A 16x4):
//   lane L: m = L&15, half = L>>4
//   A vgpr0/1 = A[m][k + 2*half + {0,1}]       (contiguous pair -> v2f load)
//   B vgpr0/1 = B[k + 2*half + {0,1}][n0 + m]  (row striped across lanes)
//   C/D vgpr r = D[r + 8*half][n0 + m]
// ---------------------------------------------------------------------------
__global__ void gemm_embW(const float* __restrict__ emb,
                          const float* __restrict__ W,
                          const float* __restrict__ bias,
                          float* __restrict__ embW,
                          int vocab) {
  const int lane   = threadIdx.x & 31;
  const int waveId = blockIdx.x * (blockDim.x >> 5) + (threadIdx.x >> 5);
  const int ntiles = ENC_DIM / 16;              // 16 N-tiles
  const int tileM  = waveId / ntiles;
  const int tileN  = waveId % ntiles;
  const int m0 = tileM * 16;
  const int n0 = tileN * 16;
  if (m0 >= vocab) return;                      // uniform per wave

  const int half = lane >> 4;
  const int mr   = lane & 15;

  int arow = m0 + mr;
  if (arow >= vocab) arow = vocab - 1;          // clamp (store guarded below)
  const float* __restrict__ Arow = emb + (size_t)arow * EMB_DIM;

  v8f acc = {};
  for (int k = 0; k < EMB_DIM; k += 4) {
    v2f a = *(const v2f*)(Arow + k + 2 * half);
    v2f bm;
    bm.x = W[(size_t)(k + 2 * half)     * ENC_DIM + n0 + mr];
    bm.y = W[(size_t)(k + 2 * half + 1) * ENC_DIM + n0 + mr];
    // 8 args: (neg_a, A, neg_b, B, c_mod, C, reuse_a, reuse_b)
    acc = __builtin_amdgcn_wmma_f32_16x16x4_f32(
        false, a, false, bm, (short)0, acc, false, false);
  }

  const float bb = bias[n0 + mr];               // per-column bias
  float* __restrict__ out = embW + (size_t)(m0 + 8 * half) * ENC_DIM + n0 + mr;

  if (m0 + 16 <= vocab) {
    // Full tile (always true when vocab % 16 == 0): wave-uniform fast path,
    // 8 unguarded stores -> no per-lane exec-mask dance, clause-friendly.
#pragma unroll
    for (int r = 0; r < 8; ++r)
      out[(size_t)r * ENC_DIM] = acc[r] + bb;
  } else {
    // Rare ragged tail: per-lane guarded stores.
#pragma unroll
    for (int r = 0; r < 8; ++r) {
      const int row = m0 + r + 8 * half;
      if (row < vocab)
        embW[(size_t)row * ENC_DIM + n0 + mr] = acc[r] + bb;
    }
  }
}

// ---------------------------------------------------------------------------
// Kernel 2: per-block column sums of the gathered rows.
// blockTot[b][t] = sum over rows i in [b*R, (b+1)*R) of embW[ids[i]][t]
// ---------------------------------------------------------------------------
__global__ void block_sums(const float* __restrict__ embW,
                           const int* __restrict__ ids,
                           float* __restrict__ blockTot,
                           int n) {
  __shared__ int sid[256];
  const int t    = threadIdx.x;
  const int base = blockIdx.x * ROWS_PER_BLK;
  float sum = 0.f;
  for (int c = 0; c < ROWS_PER_BLK; c += 256) {
    const int i = base + c + t;
    sid[t] = (i < n) ? ids[i] : 0;
    if (i + 256 < n) __builtin_prefetch(ids + i + 256, 0, 0);  // global_prefetch_b8
    __syncthreads();
    const int lim = min(256, n - (base + c));
    for (int r = 0; r < lim; ++r)
      sum += embW[(size_t)sid[r] * ENC_DIM + t];
    __syncthreads();
  }
  blockTot[(size_t)blockIdx.x * ENC_DIM + t] = sum;
}

// ---------------------------------------------------------------------------
// Kernel 3: sequential exclusive scan of block totals (196 blocks -> trivial).
// In-place: blockTot becomes per-block exclusive column offsets.
// ---------------------------------------------------------------------------
__global__ void scan_blocks(float* __restrict__ blockTot, int nblk) {
  const int t = threadIdx.x;
  float run = 0.f;
  for (int b = 0; b < nblk; ++b) {
    const float v = blockTot[(size_t)b * ENC_DIM + t];
    blockTot[(size_t)b * ENC_DIM + t] = run;
    run += v;
  }
}

// ---------------------------------------------------------------------------
// Kernel 4: emit the exclusive prefix S[0..n], S[i][t] = sum_{j<i} c_j[t].
// ---------------------------------------------------------------------------
__global__ void emit_S(const float* __restrict__ embW,
                       const int* __restrict__ ids,
                       const float* __restrict__ blockOff,
                       float* __restrict__ S,
                       int n, int nblk) {
  __shared__ int sid[256];
  const int t    = threadIdx.x;
  const int base = blockIdx.x * ROWS_PER_BLK;
  float run = blockOff[(size_t)blockIdx.x * ENC_DIM + t];
  for (int c = 0; c < ROWS_PER_BLK; c += 256) {
    const int i = base + c + t;
    sid[t] = (i < n) ? ids[i] : 0;
    __syncthreads();
    const int lim = min(256, n - (base + c));
    for (int r = 0; r < lim; ++r) {
      const int row = base + c + r;
      S[(size_t)row * ENC_DIM + t] = run;
      run += embW[(size_t)sid[r] * ENC_DIM + t];
    }
    __syncthreads();
  }
  if (blockIdx.x == nblk - 1)                   // total sum -> S[n]
    S[(size_t)n * ENC_DIM + t] = run;
}

// ---------------------------------------------------------------------------
// Kernel 5: per-block partial column max of h_i = S[i+sz_i] - S[i].
// ---------------------------------------------------------------------------
__global__ void hmax_partial(const float* __restrict__ S,
                             const int* __restrict__ sz,
                             float* __restrict__ pmax,
                             int n) {
  const int t = threadIdx.x;
  float mx = -__builtin_inff();
  for (int i = blockIdx.x; i < n; i += gridDim.x) {
    const int s  = sz[i];                       // uniform across block
    const float h = S[(size_t)(i + s) * ENC_DIM + t]
                  - S[(size_t)i       * ENC_DIM + t];
    mx = fmaxf(mx, h);
  }
  pmax[(size_t)blockIdx.x * ENC_DIM + t] = mx;
}

// ---------------------------------------------------------------------------
// Kernel 6: final reduction over block partials -> d_out[256].
// ---------------------------------------------------------------------------
__global__ void final_max(const float* __restrict__ pmax,
                          float* __restrict__ out, int nblk) {
  const int t = threadIdx.x;
  float mx = -__builtin_inff();
  for (int b = 0; b < nblk; ++b)
    mx = fmaxf(mx, pmax[(size_t)b * ENC_DIM + t]);
  out[t] = mx;
}

// ---------------------------------------------------------------------------
// Inputs (setup_inputs order): node_ids[int n], subtree_size[int n],
//   emb[f32 vocab*256], W[f32 256*256], b[f32 256]. Output: f32[256].
// ---------------------------------------------------------------------------
extern "C" void kernel_launch(void* const* d_in, const int* in_sizes, int n_in,
                              void* d_out, int out_size, void* d_ws, size_t ws_size,
                              hipStream_t stream) {
  const int*   node_ids = (const int*)d_in[0];
  const int*   subtree  = (const int*)d_in[1];
  const float* emb      = (const float*)d_in[2];
  const float* W        = (const float*)d_in[3];
  const float* bias     = (const float*)d_in[4];

  const int n     = in_sizes[0];
  const int vocab = in_sizes[2] / EMB_DIM;
  const int nblk  = (n + ROWS_PER_BLK - 1) / ROWS_PER_BLK;

  // Workspace carve-up (floats): embW | S | blockTot | pmax
  float* ws       = (float*)d_ws;
  float* embW     = ws;
  float* S        = embW + (size_t)vocab * ENC_DIM;
  float* blockTot = S + (size_t)(n + 1) * ENC_DIM;
  float* pmax     = blockTot + (size_t)nblk * ENC_DIM;

  // 1) vocab-side GEMM via f32 WMMA: 625*16 = 10,000 waves (8 waves/block).
  const int mtiles     = (vocab + 15) / 16;
  const int totalWaves = mtiles * (ENC_DIM / 16);
  const int gemmBlocks = (totalWaves + 7) / 8;
  gemm_embW<<<gemmBlocks, 256, 0, stream>>>(emb, W, bias, embW, vocab);

  // 2..4) three-pass global exclusive prefix sum over gathered rows.
  block_sums<<<nblk, 256, 0, stream>>>(embW, node_ids, blockTot, n);
  scan_blocks<<<1, 256, 0, stream>>>(blockTot, nblk);
  emit_S<<<nblk, 256, 0, stream>>>(embW, node_ids, blockTot, S, n, nblk);

  // 5..6) subtree-range diff + column max reduction.
  hmax_partial<<<MAX_GRID, 256, 0, stream>>>(S, subtree, pmax, n);
  final_max<<<1, 256, 0, stream>>>(pmax, (float*)d_out, MAX_GRID);
}